// GroupPointNet_60610578481759
// MI455X (gfx1250) — compile-verified
//
#include <hip/hip_runtime.h>
#include <stdint.h>

typedef __attribute__((ext_vector_type(2))) float v2f;
typedef __attribute__((ext_vector_type(8))) float v8f;

#define NPTS   8192
#define BATCH  4
#define MPTS   2048          // NPTS * 0.25
#define KNNK   20
#define CH     64
#define PIX    (BATCH*MPTS*KNNK)   // 163840 pixels
#define SLOPE  0.2f
#define BN_EPS 1e-5f

// ---------------------------------------------------------------------------
// Zero the BN statistics (6 arrays of 64 floats: sum/sq for 3 layers)
// ---------------------------------------------------------------------------
__global__ void zero_stats_kernel(float* s) {
    int i = blockIdx.x * 256 + threadIdx.x;
    if (i < 6 * CH) s[i] = 0.0f;
}

// ---------------------------------------------------------------------------
// Furthest point sampling: one workgroup (1024 thr = 32 waves) per batch.
// Each thread keeps 8 points + running min-dist in registers.
// ---------------------------------------------------------------------------
__global__ __launch_bounds__(1024) void fps_kernel(const float* __restrict__ p,
                                                   int* __restrict__ outIdx) {
    const int b   = blockIdx.x;
    const int tid = threadIdx.x;
    const float* pb = p + (size_t)b * NPTS * 3;

    float px[8], py[8], pz[8], dist[8];
#pragma unroll
    for (int s = 0; s < 8; ++s) {
        int i = tid + s * 1024;
        px[s] = pb[i * 3 + 0];
        py[s] = pb[i * 3 + 1];
        pz[s] = pb[i * 3 + 2];
        dist[s] = 1e10f;
    }

    __shared__ float lastx, lasty, lastz;
    __shared__ int   curIdx;
    __shared__ float redV[32];
    __shared__ int   redI[32];
    if (tid == 0) curIdx = 0;
    __syncthreads();

    for (int t = 0; t < MPTS; ++t) {
        if (tid == 0) {
            int c = curIdx;
            outIdx[b * MPTS + t] = c;
            lastx = pb[c * 3 + 0];
            lasty = pb[c * 3 + 1];
            lastz = pb[c * 3 + 2];
        }
        __syncthreads();
        float lx = lastx, ly = lasty, lz = lastz;

        float bestV = -1.0f;
        int   bestI = 0;
#pragma unroll
        for (int s = 0; s < 8; ++s) {
            float dx = px[s] - lx, dy = py[s] - ly, dz = pz[s] - lz;
            float d  = dx * dx + dy * dy + dz * dz;
            float nd = fminf(dist[s], d);
            dist[s]  = nd;
            if (nd > bestV) { bestV = nd; bestI = tid + s * 1024; }  // s ascending -> lowest idx kept
        }
        // wave32 argmax (lowest-index tiebreak)
#pragma unroll
        for (int off = 16; off > 0; off >>= 1) {
            float ov = __shfl_xor(bestV, off, 32);
            int   oi = __shfl_xor(bestI, off, 32);
            if (ov > bestV || (ov == bestV && oi < bestI)) { bestV = ov; bestI = oi; }
        }
        if ((tid & 31) == 0) { redV[tid >> 5] = bestV; redI[tid >> 5] = bestI; }
        __syncthreads();
        if (tid < 32) {
            bestV = redV[tid];
            bestI = redI[tid];
#pragma unroll
            for (int off = 16; off > 0; off >>= 1) {
                float ov = __shfl_xor(bestV, off, 32);
                int   oi = __shfl_xor(bestI, off, 32);
                if (ov > bestV || (ov == bestV && oi < bestI)) { bestV = ov; bestI = oi; }
            }
            if (tid == 0) curIdx = bestI;
        }
        __syncthreads();
    }
}

// ---------------------------------------------------------------------------
// KNN: one thread per query; points staged through LDS in 1024-pt tiles;
// branchless 20-entry sorted insert fully unrolled (stays in VGPRs).
// ---------------------------------------------------------------------------
__global__ __launch_bounds__(256) void knn_kernel(const float* __restrict__ p,
                                                  const int* __restrict__ fpsIdx,
                                                  int* __restrict__ nidx) {
    const int b = blockIdx.x / (MPTS / 256);
    const int m = (blockIdx.x % (MPTS / 256)) * 256 + threadIdx.x;
    const float* pb = p + (size_t)b * NPTS * 3;

    int qi = fpsIdx[b * MPTS + m];
    float qx = pb[qi * 3 + 0], qy = pb[qi * 3 + 1], qz = pb[qi * 3 + 2];

    float bd[KNNK];
    int   bi[KNNK];
#pragma unroll
    for (int i = 0; i < KNNK; ++i) { bd[i] = 3.0e38f; bi[i] = 0; }

    __shared__ float sp[1024 * 3];
    for (int c0 = 0; c0 < NPTS; c0 += 1024) {
        __syncthreads();
        for (int j = threadIdx.x; j < 1024 * 3; j += 256) sp[j] = pb[c0 * 3 + j];
        __syncthreads();
        for (int j = 0; j < 1024; ++j) {
            float dx = sp[j * 3 + 0] - qx;
            float dy = sp[j * 3 + 1] - qy;
            float dz = sp[j * 3 + 2] - qz;
            float d  = dx * dx + dy * dy + dz * dz;
            if (d < bd[KNNK - 1]) {
                int jj = c0 + j;
                // new[i]=old[i] if old[i]<=d; else old[i-1] if old[i-1]>d; else d
#pragma unroll
                for (int i = KNNK - 1; i >= 1; --i) {
                    bool keep  = bd[i] <= d;
                    bool shift = bd[i - 1] > d;
                    float nb = keep ? bd[i] : (shift ? bd[i - 1] : d);
                    int   ni = keep ? bi[i] : (shift ? bi[i - 1] : jj);
                    bd[i] = nb; bi[i] = ni;
                }
                if (bd[0] > d) { bd[0] = d; bi[0] = jj; }
            }
        }
    }
    int base = (b * MPTS + m) * KNNK;
#pragma unroll
    for (int i = 0; i < KNNK; ++i) nidx[base + i] = bi[i];
}

// ---------------------------------------------------------------------------
// Layer 1: gather + relative coords -> 6ch feature -> 64ch conv -> leakyrelu.
// Writes channel-major X1[c][pixel]; BN stats via LDS then global atomics.
// ---------------------------------------------------------------------------
__global__ __launch_bounds__(256) void conv1_kernel(const float* __restrict__ p,
                                                    const int* __restrict__ fpsIdx,
                                                    const int* __restrict__ nidx,
                                                    const float* __restrict__ W1,
                                                    float* __restrict__ X1,
                                                    float* __restrict__ stats) {
    int t   = blockIdx.x * 256 + threadIdx.x;      // pixel id
    int b   = t / (MPTS * KNNK);
    int rem = t % (MPTS * KNNK);
    int m   = rem / KNNK;
    const float* pb = p + (size_t)b * NPTS * 3;

    int j = nidx[t];
    int c = fpsIdx[b * MPTS + m];
    float gx = pb[j * 3 + 0], gy = pb[j * 3 + 1], gz = pb[j * 3 + 2];
    float cx = pb[c * 3 + 0], cy = pb[c * 3 + 1], cz = pb[c * 3 + 2];
    float f0 = gx - cx, f1 = gy - cy, f2 = gz - cz, f3 = gx, f4 = gy, f5 = gz;

    __shared__ float sW[CH * 6];
    __shared__ float sSum[CH], sSq[CH];
    for (int i = threadIdx.x; i < CH * 6; i += 256) sW[i] = W1[i];
    for (int i = threadIdx.x; i < CH; i += 256) { sSum[i] = 0.0f; sSq[i] = 0.0f; }
    __syncthreads();

#pragma unroll 4
    for (int o = 0; o < CH; ++o) {
        const float* w = &sW[o * 6];
        float v = w[0]*f0 + w[1]*f1 + w[2]*f2 + w[3]*f3 + w[4]*f4 + w[5]*f5;
        v = v >= 0.0f ? v : SLOPE * v;
        X1[(size_t)o * PIX + t] = v;
        atomicAdd(&sSum[o], v);
        atomicAdd(&sSq[o], v * v);
    }
    __syncthreads();
    for (int i = threadIdx.x; i < CH; i += 256) {
        atomicAdd(&stats[i], sSum[i]);
        atomicAdd(&stats[CH + i], sSq[i]);
    }
}

// ---------------------------------------------------------------------------
// stats -> affine BN params:  y = a*x + c0
// ---------------------------------------------------------------------------
__global__ void bn_finalize_kernel(const float* __restrict__ stats,
                                   const float* __restrict__ g,
                                   const float* __restrict__ bta,
                                   float* __restrict__ ac) {
    int i = threadIdx.x;
    float cnt  = (float)PIX;
    float mean = stats[i] / cnt;
    float var  = stats[CH + i] / cnt - mean * mean;
    float a    = g[i] * rsqrtf(var + BN_EPS);
    ac[i]      = a;
    ac[CH + i] = bta[i] - a * mean;
}

// ---------------------------------------------------------------------------
// 64x64 1x1 conv via V_WMMA_F32_16X16X4_F32.
// The 64x16 input tile is staged into LDS ONCE per block with
// global_load_async_to_lds_b128 (ASYNCcnt) and shared by all 4 waves,
// removing the 4x redundant L2 read. Previous layer's BN affine is fused
// into the B-fragment load from LDS.
// ---------------------------------------------------------------------------
__global__ __launch_bounds__(128) void conv_wmma_kernel(const float* __restrict__ Xin,
                                                        const float* __restrict__ W,
                                                        const float* __restrict__ ac,
                                                        float* __restrict__ Xout,
                                                        float* __restrict__ stats) {
    __shared__ float sA[CH], sC[CH];
    __shared__ float tileBuf[CH * 16];          // 4KB staging tile: [row=in-ch][col=pixel]
    for (int i = threadIdx.x; i < CH; i += 128) { sA[i] = ac[i]; sC[i] = ac[CH + i]; }
    __syncthreads();

    const int lane    = threadIdx.x & 31;
    const int wave    = threadIdx.x >> 5;   // out-channel tile 0..3
    const int halfSel = lane >> 4;          // 0 / 1
    const int l16     = lane & 15;

    // Preload weight (A) fragments: 16x4 tile per K-chunk.
    // lanes 0-15: rows M=l16, K = 4kc+{0,1}; lanes 16-31: K = 4kc+{2,3}
    v2f afrag[16];
    const int arow = wave * 16 + l16;
#pragma unroll
    for (int kc = 0; kc < 16; ++kc) {
        int col = kc * 4 + halfSel * 2;
        v2f a;
        a.x = W[arow * CH + col];
        a.y = W[arow * CH + col + 1];
        afrag[kc] = a;
    }

    float ssum[8], ssq[8];
#pragma unroll
    for (int r = 0; r < 8; ++r) { ssum[r] = 0.0f; ssq[r] = 0.0f; }

    // Staging assignment: thread t copies 16B (4 floats) of rows srow and srow+32.
    const int t128 = threadIdx.x;
    const int srow = t128 >> 2;             // 0..31
    const int scol = (t128 & 3) * 4;        // 0,4,8,12
    uint32_t ldsA0 = (uint32_t)(uintptr_t)(&tileBuf[srow * 16 + scol]);
    uint32_t ldsA1 = (uint32_t)(uintptr_t)(&tileBuf[(srow + 32) * 16 + scol]);

    const int numTiles = PIX / 16;   // 10240; 640 blocks -> uniform 16 iters/block
    for (int tile = blockIdx.x; tile < numTiles; tile += gridDim.x) {
        int p0 = tile * 16;
        // ---- async stage 64x16 f32 tile into LDS (all addresses 16B aligned) ----
        uint32_t g0 = (uint32_t)(((uint32_t)srow        * PIX + (uint32_t)(p0 + scol)) * 4u);
        uint32_t g1 = (uint32_t)(((uint32_t)(srow + 32) * PIX + (uint32_t)(p0 + scol)) * 4u);
        asm volatile("global_load_async_to_lds_b128 %0, %1, %2"
                     :: "v"(ldsA0), "v"(g0), "s"(Xin) : "memory");
        asm volatile("global_load_async_to_lds_b128 %0, %1, %2"
                     :: "v"(ldsA1), "v"(g1), "s"(Xin) : "memory");
        asm volatile("s_wait_asynccnt 0" ::: "memory");
        __syncthreads();                     // tile visible to all 4 waves

        v8f acc = {0.f, 0.f, 0.f, 0.f, 0.f, 0.f, 0.f, 0.f};
#pragma unroll
        for (int kc = 0; kc < 16; ++kc) {
            int r0 = kc * 4 + halfSel * 2;
            float x0 = tileBuf[r0 * 16 + l16];
            float x1 = tileBuf[(r0 + 1) * 16 + l16];
            v2f bfrag;
            bfrag.x = sA[r0]     * x0 + sC[r0];
            bfrag.y = sA[r0 + 1] * x1 + sC[r0 + 1];
            acc = __builtin_amdgcn_wmma_f32_16x16x4_f32(
                false, afrag[kc], false, bfrag, (short)0, acc, false, false);
        }
        // leaky relu, store (channel-major), accumulate stats
#pragma unroll
        for (int r = 0; r < 8; ++r) {
            float v = acc[r];
            v = v >= 0.0f ? v : SLOPE * v;
            int ch = wave * 16 + r + halfSel * 8;       // C/D layout: lanes>=16 -> M+8
            Xout[(size_t)ch * PIX + p0 + l16] = v;
            ssum[r] += v;
            ssq[r]  += v * v;
        }
        __syncthreads();                     // protect tileBuf before next copies
    }

    // reduce each 16-lane half independently (xor masks < 16 keep halves separate)
#pragma unroll
    for (int r = 0; r < 8; ++r) {
        float s = ssum[r], q = ssq[r];
#pragma unroll
        for (int off = 1; off < 16; off <<= 1) {
            s += __shfl_xor(s, off, 32);
            q += __shfl_xor(q, off, 32);
        }
        if (l16 == 0) {
            int ch = wave * 16 + r + halfSel * 8;
            atomicAdd(&stats[ch], s);
            atomicAdd(&stats[CH + ch], q);
        }
    }
}

// ---------------------------------------------------------------------------
// max over K (fused with final BN affine; handles a<0 via min track)
// ---------------------------------------------------------------------------
__global__ __launch_bounds__(256) void maxpool_kernel(const float* __restrict__ X3,
                                                      const float* __restrict__ ac,
                                                      float* __restrict__ out) {
    int t   = blockIdx.x * 256 + threadIdx.x;   // == (b*CH + o)*MPTS + m
    int b   = t / (CH * MPTS);
    int rem = t % (CH * MPTS);
    int o   = rem / MPTS;
    int m   = rem % MPTS;
    const float* src = X3 + (size_t)o * PIX + (size_t)(b * MPTS + m) * KNNK;
    float mx = -3.0e38f, mn = 3.0e38f;
#pragma unroll
    for (int k = 0; k < KNNK; ++k) {
        float v = src[k];
        mx = fmaxf(mx, v);
        mn = fminf(mn, v);
    }
    float a = ac[o], c0 = ac[CH + o];
    out[t] = a >= 0.0f ? (a * mx + c0) : (a * mn + c0);
}

// ---------------------------------------------------------------------------
extern "C" void kernel_launch(void* const* d_in, const int* in_sizes, int n_in,
                              void* d_out, int out_size, void* d_ws, size_t ws_size,
                              hipStream_t stream) {
    const float* p  = (const float*)d_in[0];
    const float* W1 = (const float*)d_in[1];
    const float* g1 = (const float*)d_in[2];
    const float* b1 = (const float*)d_in[3];
    const float* W2 = (const float*)d_in[4];
    const float* g2 = (const float*)d_in[5];
    const float* b2 = (const float*)d_in[6];
    const float* W3 = (const float*)d_in[7];
    const float* g3 = (const float*)d_in[8];
    const float* b3 = (const float*)d_in[9];
    float* out = (float*)d_out;

    char* ws = (char*)d_ws;
    int*   fpsIdx = (int*)(ws);                                   // 8192 int
    int*   nidx   = (int*)(ws + 32768);                           // 163840 int
    float* stats  = (float*)(ws + 32768 + PIX * 4);               // 6*64 f32
    float* acp    = stats + 6 * CH;                               // 6*64 f32
    size_t xoff   = 1u << 20;                                     // 1MB aligned
    float* X1     = (float*)(ws + xoff);                          // 64*PIX f32
    float* X2     = (float*)(ws + xoff + (size_t)CH * PIX * 4);   // 64*PIX f32
    float* X3     = X1;                                           // reuse X1

    float* stats1 = stats;            float* ac1 = acp;
    float* stats2 = stats + 2 * CH;   float* ac2 = acp + 2 * CH;
    float* stats3 = stats + 4 * CH;   float* ac3 = acp + 4 * CH;

    zero_stats_kernel<<<2, 256, 0, stream>>>(stats);
    fps_kernel<<<BATCH, 1024, 0, stream>>>(p, fpsIdx);
    knn_kernel<<<BATCH * (MPTS / 256), 256, 0, stream>>>(p, fpsIdx, nidx);
    conv1_kernel<<<PIX / 256, 256, 0, stream>>>(p, fpsIdx, nidx, W1, X1, stats1);
    bn_finalize_kernel<<<1, CH, 0, stream>>>(stats1, g1, b1, ac1);
    conv_wmma_kernel<<<640, 128, 0, stream>>>(X1, W2, ac1, X2, stats2);
    bn_finalize_kernel<<<1, CH, 0, stream>>>(stats2, g2, b2, ac2);
    conv_wmma_kernel<<<640, 128, 0, stream>>>(X2, W3, ac2, X3, stats3);
    bn_finalize_kernel<<<1, CH, 0, stream>>>(stats3, g3, b3, ac3);
    maxpool_kernel<<<(BATCH * CH * MPTS) / 256, 256, 0, stream>>>(X3, ac3, out);
}